// RPNTrainer_42494406427381
// MI455X (gfx1250) — compile-verified
//
#include <hip/hip_runtime.h>
#include <math.h>

typedef float v2f __attribute__((ext_vector_type(2)));
typedef float v8f __attribute__((ext_vector_type(8)));

// ---------------------------------------------------------------------------
// Wave32 sum-reduction through the CDNA5 matrix pipe (pure f32, no downcast).
// A-matrix (16x4 f32, 2 VGPRs): lane L<16 supplies A[L][0]=x_L, A[L][1]=0;
// lane L>=16 supplies A[L-16][2]=x_L, A[L-16][3]=0. B = all-ones 4x16
// (layout-independent). D[m,n] = x_m + x_{m+16}; per-lane sum of the 8 C/D
// VGPRs covers m=0..7 (lanes 0..15) or m=8..15 (lanes 16..31); an xor-16
// shuffle completes the sum, replicated in every lane.
// ---------------------------------------------------------------------------
__device__ __forceinline__ float wave_sum_wmma(float x) {
  v2f a; a[0] = x;    a[1] = 0.0f;
  v2f b; b[0] = 1.0f; b[1] = 1.0f;
  v8f c = {};
  c = __builtin_amdgcn_wmma_f32_16x16x4_f32(
      /*neg_a=*/false, a, /*neg_b=*/false, b,
      /*c_mod=*/(short)0, c, /*reuse_a=*/false, /*reuse_b=*/false);
  float s = c[0] + c[1] + c[2] + c[3] + c[4] + c[5] + c[6] + c[7];
  s += __shfl_xor(s, 16, 32);
  return s;
}

#define MAX_BT 512  // B*T = 256 for this problem; headroom kept

__global__ void rpn_init_ws(float* __restrict__ ws) {
  if (threadIdx.x < 3) ws[threadIdx.x] = 0.0f;
}

__global__ void rpn_loss_kernel(const float* __restrict__ reg,      // [B,A,4]
                                const float* __restrict__ cls,      // [B,A]
                                const float* __restrict__ anchors,  // [A,4]
                                const float* __restrict__ targets,  // [B,T,4]
                                float* __restrict__ ws,             // bce,sl1,npos
                                int A, int B, int T) {
  __shared__ float4 s_box[MAX_BT];  // one ds_load_b128 per inner iteration
  __shared__ float  s_ar[MAX_BT];   // precomputed target areas

  const int BT = B * T;
  const float4* tgt4 = (const float4*)targets;
  for (int i = threadIdx.x; i < BT; i += blockDim.x) {
    const float4 t = tgt4[i];
    s_box[i] = t;
    s_ar[i]  = (t.z - t.x) * (t.w - t.y);
  }
  __syncthreads();

  float bce_p = 0.0f, sl1_p = 0.0f, np_p = 0.0f;

  const float4* anc4 = (const float4*)anchors;
  const float4* reg4 = (const float4*)reg;

  const int stride = gridDim.x * blockDim.x;
  for (int a = blockIdx.x * blockDim.x + threadIdx.x; a < A; a += stride) {
    const float4 an = anc4[a];
    const float area_a = (an.z - an.x) * (an.w - an.y);

    for (int b = 0; b < B; ++b) {
      // Division-free max/argmax of IoU over T targets.
      // IoU_j = inter_j / den_j with den_j = area_a + area_t_j - inter_j > 0.
      // Compare fractions by cross-multiplication; (-1,1) init guarantees the
      // first target wins the first strict compare (jnp.argmax tie-break).
      float b_num = -1.0f, b_den = 1.0f;
      int   bj = 0;
      const int base = b * T;
      #pragma unroll 8
      for (int t = 0; t < T; ++t) {
        const int j = base + t;
        const float4 tb = s_box[j];
        const float  at = s_ar[j];
        const float iw = fmaxf(fminf(tb.z, an.z) - fmaxf(tb.x, an.x), 0.0f);
        const float ih = fmaxf(fminf(tb.w, an.w) - fmaxf(tb.y, an.y), 0.0f);
        const float inter = iw * ih;
        const float den   = (area_a + at) - inter;
        const bool  gt    = inter * b_den > b_num * den;
        b_num = gt ? inter : b_num;
        b_den = gt ? den   : b_den;
        bj    = gt ? j     : bj;
      }

      const float cv   = cls[(size_t)b * A + a];
      // max_iou > 0.5  <=>  2*b_num > b_den  (b_den > 0)
      const float posf = (b_num + b_num > b_den) ? 1.0f : 0.0f;
      np_p += posf;
      // stable softplus(cv) - cv*posf
      bce_p += fmaxf(cv, 0.0f) + log1pf(expf(-fabsf(cv))) - cv * posf;

      if (posf > 0.0f) {
        const float4 rg = reg4[(size_t)b * A + a];
        const float4 tb = s_box[bj];
        const float e0 = rg.x - (tb.x - an.x);
        const float e1 = rg.y - (tb.y - an.y);
        const float e2 = rg.z - (tb.z - an.z);
        const float e3 = rg.w - (tb.w - an.w);
        const float er[4] = { e0, e1, e2, e3 };
        #pragma unroll
        for (int k = 0; k < 4; ++k) {
          const float ae = fabsf(er[k]);
          sl1_p += (ae < 1.0f) ? (0.5f * er[k] * er[k]) : (ae - 0.5f);
        }
      }
    }
  }

  // Fully converged here: EXEC all-1s for WMMA (ISA 7.12 requirement).
  const float bce_w = wave_sum_wmma(bce_p);
  const float sl1_w = wave_sum_wmma(sl1_p);
  const float np_w  = wave_sum_wmma(np_p);

  if ((threadIdx.x & 31) == 0) {
    atomicAdd(&ws[0], bce_w);
    atomicAdd(&ws[1], sl1_w);
    atomicAdd(&ws[2], np_w);
  }
}

__global__ void rpn_finalize(const float* __restrict__ ws,
                             float* __restrict__ out, int BA) {
  const float bce = ws[0];
  const float sl1 = ws[1];
  const float np  = ws[2];
  out[0] = bce / (float)BA;                                     // cls_loss
  const float r = (np > 0.0f) ? (sl1 / fmaxf(np, 1.0f)) : 0.0f;
  out[1] = r * 0.25f;                                           // reg_loss / 4
}

extern "C" void kernel_launch(void* const* d_in, const int* in_sizes, int n_in,
                              void* d_out, int out_size, void* d_ws, size_t ws_size,
                              hipStream_t stream) {
  const float* reg     = (const float*)d_in[0];  // [B,A,4]
  const float* cls     = (const float*)d_in[1];  // [B,A]
  const float* anchors = (const float*)d_in[2];  // [A,4]
  const float* targets = (const float*)d_in[3];  // [B,T,4]

  const int A = in_sizes[2] / 4;
  const int B = in_sizes[1] / A;
  const int T = in_sizes[3] / (4 * B);

  float* ws = (float*)d_ws;

  rpn_init_ws<<<1, 64, 0, stream>>>(ws);

  const int threads = 256;                 // 8 wave32 per block
  int blocks = (A + threads - 1) / threads;
  if (blocks > 4096) blocks = 4096;
  rpn_loss_kernel<<<blocks, threads, 0, stream>>>(reg, cls, anchors, targets,
                                                  ws, A, B, T);

  rpn_finalize<<<1, 1, 0, stream>>>(ws, (float*)d_out, A * B);
}